// GCN_55559696941219
// MI455X (gfx1250) — compile-verified
//
#include <hip/hip_runtime.h>
#include <math.h>

typedef __attribute__((ext_vector_type(2))) float v2f;
typedef __attribute__((ext_vector_type(8))) float v8f;

#define NFEAT 128
#define TPB   256

// ---------- hardware f32 atomic add (global_atomic_add_f32, no CAS loop) ----
__device__ __forceinline__ void atomAddF(float* p, float v) {
    unsafeAtomicAdd(p, v);
}

// ---------------------- degree / normalization ------------------------------
__global__ void k_init_deg(float* __restrict__ deg, int n) {
    int i = blockIdx.x * blockDim.x + threadIdx.x;
    if (i < n) deg[i] = 1.0f;                 // self-loop contribution
}

__global__ void k_edge_deg(const int* __restrict__ dst, float* __restrict__ deg, int e) {
    int i = blockIdx.x * blockDim.x + threadIdx.x;
    if (i < e) atomAddF(&deg[dst[i]], 1.0f);
}

__global__ void k_dinv(float* __restrict__ deg, int n) {
    int i = blockIdx.x * blockDim.x + threadIdx.x;
    if (i < n) deg[i] = rsqrtf(deg[i]);       // deg >= 1 always (self-loops)
}

// ---------------------- WMMA f32 GEMM: out[M,128] = A[M,128] @ W[128,128] ---
// One wave computes one 16x16 tile with V_WMMA_F32_16X16X4_F32 (K-step 4).
// blockDim = 256 (8 waves); wave w handles column tile w; blockIdx.x = row tile.
// A layout (16x4 f32): lane = (k>=2?16:0)+m, vgpr = k%2  -> contiguous float2.
// B layout (4x16 f32): lane = (k>=2?16:0)+n, vgpr = k%2.
// C/D layout: vgpr r -> row (lane>=16?8:0)+r, col = lane&15.
__global__ void k_gemm_wmma(const float* __restrict__ A, const float* __restrict__ W,
                            float* __restrict__ out, int M) {
    const int lane = threadIdx.x & 31;
    const int wv   = threadIdx.x >> 5;
    const int m0   = blockIdx.x * 16;
    const int n0   = wv * 16;
    const int mrow = m0 + (lane & 15);
    const int koff = (lane >> 4) << 1;        // 0 or 2
    const int ncol = n0 + (lane & 15);

    const float* __restrict__ arow = A + (size_t)mrow * NFEAT;
    v8f acc = {0.f, 0.f, 0.f, 0.f, 0.f, 0.f, 0.f, 0.f};

#pragma unroll
    for (int k = 0; k < NFEAT; k += 4) {
        const float2 av = *(const float2*)(arow + k + koff);     // K=koff, koff+1
        v2f a; a.x = av.x; a.y = av.y;
        v2f b;
        b.x = W[(size_t)(k + koff) * NFEAT + ncol];
        b.y = W[(size_t)(k + koff + 1) * NFEAT + ncol];
        acc = __builtin_amdgcn_wmma_f32_16x16x4_f32(
                  false, a, false, b, (short)0, acc, false, false);
    }

    const int mb = m0 + ((lane >> 4) << 3);   // +0 or +8
#pragma unroll
    for (int r = 0; r < 8; ++r)
        out[(size_t)(mb + r) * NFEAT + ncol] = acc[r];
}

// --------------- agg init with self-loop: agg[v] = dinv[v]^2 * h[v] ---------
__global__ void k_selfloop(const float* __restrict__ h, const float* __restrict__ dinv,
                           float* __restrict__ agg, int n) {
    int i = blockIdx.x * blockDim.x + threadIdx.x;    // n*32 float4 chunks
    if (i >= n * 32) return;
    int v  = i >> 5;
    int c4 = (i & 31) << 2;
    float w = dinv[v] * dinv[v];
    float4 hv = *(const float4*)(h + (size_t)v * NFEAT + c4);
    float4 o;
    o.x = w * hv.x; o.y = w * hv.y; o.z = w * hv.z; o.w = w * hv.w;
    *(float4*)(agg + (size_t)v * NFEAT + c4) = o;
}

// --------------- edge scatter: agg[dst] += dinv[s]*dinv[d] * h[src] ---------
// One wave per edge; each lane handles a float4 (4 features).
__global__ void k_scatter(const float* __restrict__ h, const int* __restrict__ src,
                          const int* __restrict__ dst, const float* __restrict__ dinv,
                          float* __restrict__ agg, int e) {
    long long gid = (long long)blockIdx.x * blockDim.x + threadIdx.x;
    int wid  = (int)(gid >> 5);
    int lane = (int)(gid & 31);
    if (wid >= e) return;
    int s = src[wid], d = dst[wid];
    float w = dinv[s] * dinv[d];
    float4 hv = *(const float4*)(h + (size_t)s * NFEAT + lane * 4);
    float* a = agg + (size_t)d * NFEAT + lane * 4;
    atomAddF(a + 0, w * hv.x);
    atomAddF(a + 1, w * hv.y);
    atomAddF(a + 2, w * hv.z);
    atomAddF(a + 3, w * hv.w);
}

// --------------- bias + optional tanh, in place -----------------------------
__global__ void k_bias_act(float* __restrict__ x, const float* __restrict__ b,
                           int total, int do_tanh) {
    int i = blockIdx.x * blockDim.x + threadIdx.x;
    if (i >= total) return;
    float v = x[i] + b[i & (NFEAT - 1)];
    x[i] = do_tanh ? tanhf(v) : v;
}

// --------------- mean pool over graphs --------------------------------------
__global__ void k_zero(float* __restrict__ p, int n) {
    int i = blockIdx.x * blockDim.x + threadIdx.x;
    if (i < n) p[i] = 0.0f;
}

__global__ void k_pool(const float* __restrict__ x, const int* __restrict__ batch,
                       float* __restrict__ pooled, float* __restrict__ counts, int n) {
    long long gid = (long long)blockIdx.x * blockDim.x + threadIdx.x;
    int v    = (int)(gid >> 5);
    int lane = (int)(gid & 31);
    if (v >= n) return;
    int g = batch[v];
    float4 hv = *(const float4*)(x + (size_t)v * NFEAT + lane * 4);
    float* p = pooled + (size_t)g * NFEAT + lane * 4;
    atomAddF(p + 0, hv.x);
    atomAddF(p + 1, hv.y);
    atomAddF(p + 2, hv.z);
    atomAddF(p + 3, hv.w);
    if (lane == 0) atomAddF(&counts[g], 1.0f);
}

// --------------- head: pooled/cnt @ Wout + bout, log_softmax ----------------
__global__ void k_head(const float* __restrict__ pooled, const float* __restrict__ counts,
                       const float* __restrict__ Wout, const float* __restrict__ bout,
                       float* __restrict__ out, int G, int C) {
    int g = blockIdx.x * blockDim.x + threadIdx.x;
    if (g >= G) return;
    float inv = 1.0f / fmaxf(counts[g], 1.0f);
    float logit[10];
#pragma unroll
    for (int c = 0; c < 10; ++c) logit[c] = bout[c];
    for (int f = 0; f < NFEAT; ++f) {
        float pv = pooled[(size_t)g * NFEAT + f] * inv;
#pragma unroll
        for (int c = 0; c < 10; ++c) logit[c] += pv * Wout[f * 10 + c];
    }
    float mx = logit[0];
#pragma unroll
    for (int c = 1; c < 10; ++c) mx = fmaxf(mx, logit[c]);
    float se = 0.0f;
#pragma unroll
    for (int c = 0; c < 10; ++c) se += expf(logit[c] - mx);
    float lse = mx + logf(se);
#pragma unroll
    for (int c = 0; c < 10; ++c) out[(size_t)g * 10 + c] = logit[c] - lse;
}

// ============================================================================
extern "C" void kernel_launch(void* const* d_in, const int* in_sizes, int n_in,
                              void* d_out, int out_size, void* d_ws, size_t ws_size,
                              hipStream_t stream) {
    const float* x    = (const float*)d_in[0];
    const int*   ei   = (const int*)d_in[1];     // [2, E] flat: row0=src, row1=dst
    const int*   batch= (const int*)d_in[2];
    const float* W0   = (const float*)d_in[3];
    const float* b0   = (const float*)d_in[4];
    const float* W1   = (const float*)d_in[5];
    const float* b1   = (const float*)d_in[6];
    const float* W2   = (const float*)d_in[7];
    const float* b2   = (const float*)d_in[8];
    const float* Wout = (const float*)d_in[9];
    const float* bout = (const float*)d_in[10];
    float* out = (float*)d_out;

    const int N = in_sizes[0] / NFEAT;           // 50000
    const int E = in_sizes[1] / 2;               // 1600000
    const int G = out_size / 10;                 // 512

    const int* src = ei;
    const int* dst = ei + E;

    // workspace layout (floats)
    float* ws     = (float*)d_ws;
    float* P      = ws;                               // [N,128] agg / layer output
    float* Q      = P + (size_t)N * NFEAT;            // [N,128] gemm output h
    float* dinv   = Q + (size_t)N * NFEAT;            // [N] deg -> rsqrt(deg)
    float* pooled = dinv + N;                         // [G,128]
    float* counts = pooled + (size_t)G * NFEAT;       // [G]

    const int bN   = (N + TPB - 1) / TPB;
    const int bE   = (E + TPB - 1) / TPB;
    const int bNF  = (N * NFEAT + TPB - 1) / TPB;
    const int bN32 = (N * 32 + TPB - 1) / TPB;
    const int bE32 = (int)(((long long)E * 32 + TPB - 1) / TPB);

    // degree + symmetric normalization
    k_init_deg<<<bN, TPB, 0, stream>>>(dinv, N);
    k_edge_deg<<<bE, TPB, 0, stream>>>(dst, dinv, E);
    k_dinv   <<<bN, TPB, 0, stream>>>(dinv, N);

    // three GCN layers (tanh on first two)
    const float* Ws[3] = {W0, W1, W2};
    const float* bs[3] = {b0, b1, b2};
    for (int l = 0; l < 3; ++l) {
        const float* in = (l == 0) ? x : P;
        k_gemm_wmma<<<N / 16, TPB, 0, stream>>>(in, Ws[l], Q, N);
        k_selfloop <<<bN32, TPB, 0, stream>>>(Q, dinv, P, N);
        k_scatter  <<<bE32, TPB, 0, stream>>>(Q, src, dst, dinv, P, E);
        k_bias_act <<<bNF, TPB, 0, stream>>>(P, bs[l], N * NFEAT, (l < 2) ? 1 : 0);
    }

    // global mean pool + output head with log_softmax
    k_zero<<<(G * NFEAT + G + TPB - 1) / TPB, TPB, 0, stream>>>(pooled, G * NFEAT + G);
    k_pool<<<bN32, TPB, 0, stream>>>(P, batch, pooled, counts, N);
    k_head<<<(G + TPB - 1) / TPB, TPB, 0, stream>>>(pooled, counts, Wout, bout, out, G, 10);
}